// TensorBasisLayer_76639396430001
// MI455X (gfx1250) — compile-verified
//
#include <hip/hip_runtime.h>
#include <hip/hip_bf16.h>
#include <stdint.h>
#include <stddef.h>

// TensorBasisLayer (DimeNet/GemNet quadruplet basis) for gfx1250 / MI455X.
// S=7 spherical, R=6 radial -> 49 columns, 294 outputs per quad.
#define NS   7
#define NR   6
#define NSR  42    // NS*NR   per-edge radial row
#define NCOL 49    // NS*NS   sph columns
#define NOUT 294   // NCOL*NR outputs per quad
#define QPW  32    // quads per wave (one per lane in phase A)
#define WPB  4     // waves per block
#define QPB  (QPW * WPB)

// l(c) = floor(sqrt(c)) lookup (== _L_INDEX in the reference)
__constant__ unsigned char c_LC[NCOL] = {
  0,
  1,1,1,
  2,2,2,2,2,
  3,3,3,3,3,3,3,
  4,4,4,4,4,4,4,4,4,
  5,5,5,5,5,5,5,5,5,5,5,
  6,6,6,6,6,6,6,6,6,6,6,6,6
};

// ---------------------------------------------------------------------------
// Kernel 0: single-thread double-precision reproduction of the reference's
// _jn_zeros bisection, Bessel normalization, and per-column sph-harmonic
// coefficients.  Writes Z[42], BNORM[42], Kcol[49].
// ---------------------------------------------------------------------------
__device__ __forceinline__ double sph_jn_d(int n, double x) {
  double j0 = sin(x) / x;
  if (n == 0) return j0;
  double j1 = sin(x) / (x * x) - cos(x) / x;
  double jm1 = j0, jc = j1;
  for (int l = 1; l < n; ++l) {
    double nx = (2.0 * l + 1.0) / x * jc - jm1;
    jm1 = jc; jc = nx;
  }
  return jc;
}

__global__ void tb_init_consts(float* __restrict__ Z, float* __restrict__ BN,
                               float* __restrict__ Kc) {
  if (threadIdx.x != 0 || blockIdx.x != 0) return;
  const double PI = 3.14159265358979323846;
  double zeros[NS][NR];
  double points[NS + NR];
  for (int i = 0; i < NR; ++i) zeros[0][i] = (i + 1) * PI;
  const int np0 = NR + NS - 1;
  for (int i = 0; i < np0; ++i) points[i] = (i + 1) * PI;
  for (int order = 1; order < NS; ++order) {
    int nroots = NR + NS - 1 - order;
    double rac[NS + NR];
    for (int j = 0; j < nroots; ++j) {
      double a = points[j], b = points[j + 1];
      double fa = sph_jn_d(order, a);
      for (int it = 0; it < 80; ++it) {
        double m = 0.5 * (a + b);
        double fm = sph_jn_d(order, m);
        if (fa * fm <= 0.0) { b = m; } else { a = m; fa = fm; }
      }
      rac[j] = 0.5 * (a + b);
    }
    for (int i = 0; i < nroots; ++i) points[i] = rac[i];
    for (int i = 0; i < NR; ++i) zeros[order][i] = rac[i];
  }
  for (int l = 0; l < NS; ++l)
    for (int j = 0; j < NR; ++j) {
      double z = zeros[l][j];
      Z[l * NR + j] = (float)z;
      double jl1 = sph_jn_d(l + 1, z);
      BN[l * NR + j] = (float)(1.0 / sqrt(0.5 * jl1 * jl1));
    }
  // Per-column coefficients: K(0) for m=0 columns, sqrt(2)*K(m) otherwise,
  // in reference column order (per l: m0, cos m=1..l, sin m=l..1).
  const double SQ2 = 1.41421356237309504880;
  int cidx = 0;
  for (int l = 0; l < NS; ++l) {
    for (int pass = 0; pass < 3; ++pass) {
      int mlo = (pass == 0) ? 0 : 1;
      int mhi = (pass == 0) ? 0 : l;
      for (int mm = 0; mm <= mhi - mlo; ++mm) {
        int m = (pass == 2) ? (mhi - mm) : (mlo + mm);   // cos: 1..l, sin: l..1
        double ratio = 1.0;
        for (int t = l - m + 1; t <= l + m; ++t) ratio /= (double)t;
        double K = sqrt((2.0 * l + 1.0) / (4.0 * PI) * ratio);
        Kc[cidx++] = (float)((m == 0) ? K : SQ2 * K);
      }
      if (pass == 0 && l == 0) break;  // l=0 has only the m0 column
    }
  }
}

// ---------------------------------------------------------------------------
// Kernel 1: per-edge radial table. One thread per (edge, l*6+j).
// tab[e*42 + l*6 + j] = u(d) * BNORM[l,j] * j_l(Z[l,j]*d) * (1/cutoff)^1.5
// ---------------------------------------------------------------------------
__global__ __launch_bounds__(256) void tb_edge_rbf(
    const float* __restrict__ D, const float* __restrict__ Z,
    const float* __restrict__ BN, float* __restrict__ tab, int total) {
  int idx = blockIdx.x * 256 + threadIdx.x;
  if (idx >= total) return;
  int e = idx / NSR;
  int k = idx - e * NSR;
  int l = k / NR;
  float d  = D[e] * 0.2f;
  float d2 = d * d;
  float d5 = d2 * d2 * d;
  float u  = 1.0f / d + d5 * (-28.0f + d * (48.0f - 21.0f * d));
  if (!(d < 1.0f)) u = 0.0f;
  float x = d * Z[k];
  float s, c;
  __sincosf(x, &s, &c);
  float j0 = s / x;
  float jl = j0;
  if (l > 0) {
    float jm = j0;
    float jc = s / (x * x) - c / x;
    for (int t = 1; t < l; ++t) {
      float nx = (2.0f * t + 1.0f) / x * jc - jm;
      jm = jc; jc = nx;
    }
    jl = jc;
  }
  const float normc = 0.08944271909999159f;    // (1/5)^1.5
  tab[idx] = u * BN[k] * jl * normc;
}

// ---------------------------------------------------------------------------
// Kernel 2: lane-per-quad sph phase + async-DMA gather/product/store phase.
//
// Phase A: lane owns quad q = qbase+lane. Fully unrolled triangular Legendre
//          recurrence (28 P entries) + Chebyshev cos(m*phi)/sin(m*phi)
//          recurrence -> all 49 columns for 32 quads in ~270 wave-instrs.
// Phase B: loop over the wave's 32 quads; double-buffered
//          global_load_async_to_lds_b64 stages each 168B rbf row (21 lanes),
//          product written as coalesced float2 (256B per store instr).
// ---------------------------------------------------------------------------
__global__ __launch_bounds__(WPB * 32) void tb_quad(
    const float* __restrict__ tab, const float* __restrict__ alpha,
    const float* __restrict__ theta, const int* __restrict__ id4,
    const float* __restrict__ Kc, float* __restrict__ out, int nq) {
  __shared__ float s_sph[WPB][QPW * NCOL];   // [wave][quad*49 + c]
  __shared__ float s_rbf[WPB][2][48];        // double-buffered DMA target
  __shared__ float s_K[NCOL + 3];

  const int w    = threadIdx.x >> 5;
  const int lane = threadIdx.x & 31;

  if (threadIdx.x < NCOL) s_K[threadIdx.x] = Kc[threadIdx.x];
  __syncthreads();

  const int qbase = (blockIdx.x * WPB + w) * QPW;
  if (qbase >= nq) return;
  const int tcount = (nq - qbase < QPW) ? (nq - qbase) : QPW;

  // ---- coalesced per-lane loads (lane = quad) ----
  int qme = qbase + lane;
  if (qme >= nq) qme = nq - 1;
  const int   eL = id4[qme];
  const float al = alpha[qme];
  const float ph = theta[qme];

  // ---- kick off DMA for quad 0's rbf row ASAP ----
  {
    int e0 = __shfl(eL, 0, 32);
    if (lane < 21) {
      unsigned loff = (unsigned)(uintptr_t)(&s_rbf[w][0][0]) + (unsigned)(lane * 8);
      uint64_t ga = (uint64_t)(uintptr_t)(tab + (size_t)e0 * NSR) + (uint64_t)(lane * 8);
      asm volatile("global_load_async_to_lds_b64 %0, %1, off"
                   :: "v"(loff), "v"(ga) : "memory");
    }
  }

  // ---- Phase A: all 49 sph columns for this lane's quad ----
  {
    float ct, st;
    __sincosf(al, &st, &ct);
    float c1, s1;
    __sincosf(ph, &s1, &c1);

    // Associated Legendre triangle P[l][m], fully unrolled.
    float P[NS][NS];
    P[0][0] = 1.0f;
    #pragma unroll
    for (int m = 0; m < NS; ++m) {
      if (m > 0) P[m][m] = (1.0f - 2.0f * m) * st * P[m - 1][m - 1];
      if (m + 1 < NS) P[m + 1][m] = (2.0f * m + 1.0f) * ct * P[m][m];
      #pragma unroll
      for (int l = m + 2; l < NS; ++l)
        P[l][m] = ((2.0f * l - 1.0f) * ct * P[l - 1][m]
                   - (float)(l + m - 1) * P[l - 2][m]) * (1.0f / (float)(l - m));
    }
    // cos(m*phi), sin(m*phi) via angle-addition recurrence.
    float cm[NS], sm[NS];
    cm[0] = 1.0f; sm[0] = 0.0f;
    #pragma unroll
    for (int m = 1; m < NS; ++m) {
      cm[m] = cm[m - 1] * c1 - sm[m - 1] * s1;
      sm[m] = sm[m - 1] * c1 + cm[m - 1] * s1;
    }
    // Emit 49 columns in reference order (stride 49 is coprime with 64 banks).
    float* myrow = &s_sph[w][lane * NCOL];
    int cidx = 0;
    #pragma unroll
    for (int l = 0; l < NS; ++l) {
      myrow[cidx] = s_K[cidx] * P[l][0]; cidx++;
      #pragma unroll
      for (int m = 1; m <= l; ++m) { myrow[cidx] = s_K[cidx] * P[l][m] * cm[m]; cidx++; }
      #pragma unroll
      for (int m = l; m >= 1; --m) { myrow[cidx] = s_K[cidx] * P[l][m] * sm[m]; cidx++; }
    }
  }

  // ---- per-lane output indices, hoisted out of the quad loop ----
  // pair p covers out elements i=2p, 2p+1 (same column c since NR is even).
  int pc[5], poff[5];
  bool pvalid[5];
  #pragma unroll
  for (int k = 0; k < 5; ++k) {
    int idx2 = k * 32 + lane;            // pair index, 147 pairs total
    pvalid[k] = (idx2 < 147);
    int i = 2 * idx2;
    int c = i / NR; if (c > NCOL - 1) c = NCOL - 1;
    int j = i - c * NR;
    pc[k]   = c;
    poff[k] = (int)c_LC[c] * NR + j;
  }

  // ---- Phase B: double-buffered gather / product / store ----
  float* op = out + (size_t)qbase * NOUT;
  for (int t = 0; t < tcount; ++t) {
    if (t + 1 < tcount) {
      // reuse of buffer (t+1)&1: make sure our older ds reads drained,
      // then issue next row's DMA and wait for row t (<=1 outstanding).
      asm volatile("s_wait_dscnt 0x0" ::: "memory");
      int en = __shfl(eL, t + 1, 32);
      if (lane < 21) {
        unsigned loff = (unsigned)(uintptr_t)(&s_rbf[w][(t + 1) & 1][0])
                        + (unsigned)(lane * 8);
        uint64_t ga = (uint64_t)(uintptr_t)(tab + (size_t)en * NSR)
                      + (uint64_t)(lane * 8);
        asm volatile("global_load_async_to_lds_b64 %0, %1, off"
                     :: "v"(loff), "v"(ga) : "memory");
      }
      asm volatile("s_wait_asynccnt 0x1" ::: "memory");
    } else {
      asm volatile("s_wait_asynccnt 0x0" ::: "memory");
    }

    const float* srow = &s_rbf[w][t & 1][0];
    const float* ssph = &s_sph[w][t * NCOL];
    #pragma unroll
    for (int k = 0; k < 5; ++k) {
      if (pvalid[k]) {
        float sv = ssph[pc[k]];
        float2 v;
        v.x = sv * srow[poff[k]];
        v.y = sv * srow[poff[k] + 1];
        *(float2*)(op + (k * 64 + lane * 2)) = v;   // 8B-aligned, coalesced
      }
    }
    op += NOUT;
  }
}

// ---------------------------------------------------------------------------
// Launch: init consts -> edge table (workspace) -> quad product.
// Workspace: [0, nedge*42) rbf table | Z[42] | BNORM[42] | Kcol[49].
// ---------------------------------------------------------------------------
extern "C" void kernel_launch(void* const* d_in, const int* in_sizes, int n_in,
                              void* d_out, int out_size, void* d_ws, size_t ws_size,
                              hipStream_t stream) {
  const float* D   = (const float*)d_in[0];   // D_ca          [n_edges]
  const float* A   = (const float*)d_in[1];   // Alpha_cab     [n_quads]
  const float* T   = (const float*)d_in[2];   // Theta_cabd    [n_quads]
  const int*   id4 = (const int*)  d_in[3];   // id4_reduce_ca [n_quads]
  // d_in[4] (Kidx) is unused by the reference output.
  const int nedge = in_sizes[0];
  const int nq    = in_sizes[1];
  float* out = (float*)d_out;

  float* tab = (float*)d_ws;                  // nedge*42 floats
  float* Z   = tab + (size_t)nedge * NSR;
  float* BN  = Z + NSR;
  float* Kc  = BN + NSR;

  tb_init_consts<<<1, 1, 0, stream>>>(Z, BN, Kc);

  const int totE = nedge * NSR;
  tb_edge_rbf<<<(totE + 255) / 256, 256, 0, stream>>>(D, Z, BN, tab, totE);

  tb_quad<<<(nq + QPB - 1) / QPB, WPB * 32, 0, stream>>>(tab, A, T, id4, Kc, out, nq);
}